// HNetMaxAbs_42271068127508
// MI455X (gfx1250) — compile-verified
//
#include <hip/hip_runtime.h>
#include <hip/hip_bf16.h>
#include <stdint.h>

// Problem constants (fixed by the reference's setup_inputs()).
constexpr int NPTS  = 256;   // points
constexpr int NNODE = 1024;  // nodes (reduction dim)
constexpr int NCMP  = 1024;  // components
constexpr int TP    = 16;    // points per block tile
constexpr int TC    = 16;    // components per block tile
constexpr int KN    = 256;   // nodes per LDS chunk
constexpr int NCH   = NNODE / KN;  // 4 chunks

// Issue 8 async global->LDS b128 copies (64B of x-row + 64B of mask-row for
// this lane). INST_OFFSET is added to BOTH the memory and LDS address per the
// CDNA5 ISA pseudocode, so one (ldsoff, gaddr) pair covers 4 consecutive
// 16-byte transfers via offset: immediates. Tracked by ASYNCcnt (+8 per wave).
__device__ __forceinline__ void issue_chunk_async(const float* xg, const int* mg,
                                                  void* sx_dst, void* sm_dst) {
  // Low 32 bits of a flat shared-aperture pointer == LDS byte offset.
  uint32_t lx = (uint32_t)(uintptr_t)sx_dst;
  uint32_t lm = (uint32_t)(uintptr_t)sm_dst;
  asm volatile(
      "global_load_async_to_lds_b128 %0, %2, off\n\t"
      "global_load_async_to_lds_b128 %0, %2, off offset:16\n\t"
      "global_load_async_to_lds_b128 %0, %2, off offset:32\n\t"
      "global_load_async_to_lds_b128 %0, %2, off offset:48\n\t"
      "global_load_async_to_lds_b128 %1, %3, off\n\t"
      "global_load_async_to_lds_b128 %1, %3, off offset:16\n\t"
      "global_load_async_to_lds_b128 %1, %3, off offset:32\n\t"
      "global_load_async_to_lds_b128 %1, %3, off offset:48"
      :: "v"(lx), "v"(lm), "v"(xg), "v"(mg)
      : "memory");
}

__global__ __launch_bounds__(TP * TC) void HNetMaxAbs_42271068127508_kernel(
    const float* __restrict__ x,      // [NPTS][NNODE]
    const int* __restrict__ states,   // [NCMP][NNODE]
    float* __restrict__ out) {        // [2][NPTS][NCMP]
  __shared__ __align__(16) float sx[2][TP][KN];
  __shared__ __align__(16) int   sm[2][TC][KN];

  const int t  = threadIdx.x;
  const int pr = t >> 4;    // 0..15: point row within tile
  const int cr = t & 15;    // 0..15: component col within tile
  const int p0 = blockIdx.y * TP;
  const int c0 = blockIdx.x * TC;

  // Loader role: thread t stages 16 floats of x row (p0+pr) and 16 ints of
  // mask row (c0+pr), segment cr (16 elems = 64B each).
  const int r = pr, seg = cr;
  const float* xg = x      + (size_t)(p0 + r) * NNODE + seg * 16;
  const int*   mg = states + (size_t)(c0 + r) * NNODE + seg * 16;

  issue_chunk_async(xg, mg, &sx[0][r][seg * 16], &sm[0][r][seg * 16]);

  float best = -__builtin_inff();
  int   bidx = 0;

  for (int ch = 0; ch < NCH; ++ch) {
    const int buf = ch & 1;
    if (ch + 1 < NCH) {
      // Prefetch next chunk into the other buffer, then wait until only those
      // 8 async ops remain outstanding -> current chunk is resident in LDS.
      issue_chunk_async(xg + (ch + 1) * KN, mg + (ch + 1) * KN,
                        &sx[buf ^ 1][r][seg * 16], &sm[buf ^ 1][r][seg * 16]);
      asm volatile("s_wait_asynccnt 0x8" ::: "memory");
    } else {
      asm volatile("s_wait_asynccnt 0x0" ::: "memory");
    }
    __syncthreads();  // all waves' chunk-ch data visible in LDS

    const float* sxp = &sx[buf][pr][0];
    const int*   smc = &sm[buf][cr][0];
    const int nbase = ch * KN;
#pragma unroll 4
    for (int j = 0; j < KN; j += 4) {
      float4 xv = *reinterpret_cast<const float4*>(sxp + j);
      int4   mv = *reinterpret_cast<const int4*>(smc + j);
      float v;
      v = mv.x ? __builtin_fabsf(xv.x) : -__builtin_inff();
      if (v > best) { best = v; bidx = nbase + j + 0; }   // strict > keeps first argmax
      v = mv.y ? __builtin_fabsf(xv.y) : -__builtin_inff();
      if (v > best) { best = v; bidx = nbase + j + 1; }
      v = mv.z ? __builtin_fabsf(xv.z) : -__builtin_inff();
      if (v > best) { best = v; bidx = nbase + j + 2; }
      v = mv.w ? __builtin_fabsf(xv.w) : -__builtin_inff();
      if (v > best) { best = v; bidx = nbase + j + 3; }
    }
    __syncthreads();  // done reading buf before it is overwritten next iter
  }

  const int p = p0 + pr;
  const int c = c0 + cr;
  const bool has_edges = best > -__builtin_inff();  // |x| >= 0, so any masked node wins
  out[(size_t)p * NCMP + c]                             = has_edges ? best : 0.0f;
  out[(size_t)NPTS * NCMP + (size_t)p * NCMP + c]       = has_edges ? (float)bidx : 0.0f;
}

extern "C" void kernel_launch(void* const* d_in, const int* in_sizes, int n_in,
                              void* d_out, int out_size, void* d_ws, size_t ws_size,
                              hipStream_t stream) {
  (void)in_sizes; (void)n_in; (void)out_size; (void)d_ws; (void)ws_size;
  const float* x      = (const float*)d_in[0];
  const int*   states = (const int*)d_in[1];
  float*       out    = (float*)d_out;

  dim3 grid(NCMP / TC, NPTS / TP);  // 64 x 16 = 1024 workgroups
  dim3 block(TP * TC);              // 256 threads = 8 wave32
  HNetMaxAbs_42271068127508_kernel<<<grid, block, 0, stream>>>(x, states, out);
}